// NaturalDistanceWeighting_9002251452944
// MI455X (gfx1250) — compile-verified
//
#include <hip/hip_runtime.h>
#include <math.h>

typedef float v2f __attribute__((ext_vector_type(2)));
typedef float v8f __attribute__((ext_vector_type(8)));

#define FEAT_DIM 512
#define TRAIN_EPOCH 200
#define TOPK 9
#define EPSN 1e-12f

// ---------------------------------------------------------------------------
// Kernel 1: inverse L2 norms of feat rows (B rows) and weight rows (C rows).
// One wave (32 lanes) per row of length 512.
// ---------------------------------------------------------------------------
__global__ __launch_bounds__(256) void norms_kernel(const float* __restrict__ feat,
                                                    const float* __restrict__ weight,
                                                    float* __restrict__ invf,
                                                    float* __restrict__ invw,
                                                    int B, int C) {
  int row  = blockIdx.x * 8 + (threadIdx.x >> 5);
  int lane = threadIdx.x & 31;
  const float* src;
  float* dst;
  if (row < B) {
    src = feat + (size_t)row * FEAT_DIM;
    dst = invf + row;
  } else {
    int r = row - B;
    if (r >= C) return;
    src = weight + (size_t)r * FEAT_DIM;
    dst = invw + r;
  }
  float s = 0.f;
  for (int c = lane; c < FEAT_DIM; c += 32) {
    float v = src[c];
    s += v * v;
  }
  #pragma unroll
  for (int off = 16; off > 0; off >>= 1) s += __shfl_xor(s, off, 32);
  if (lane == 0) *dst = 1.0f / fmaxf(sqrtf(s), EPSN);
}

// ---------------------------------------------------------------------------
// Kernel 2: f32 WMMA GEMM  C[M,N] = A[M,K] * Bw[N,K]^T  (K = 512)
// Block tile 128x128x32, 256 threads = 8 waves, each wave: 2x4 tiles of 16x16.
// Uses V_WMMA_F32_16X16X4_F32 per the CDNA5 ISA 32-bit A/B VGPR layouts.
// ---------------------------------------------------------------------------
#define BM 128
#define BN 128
#define BKT 32
#define LDSS 36   // padded dword stride: 16B-aligned b128 stores, conflict-free b64 frag loads

__global__ __launch_bounds__(256) void gemm_wmma_f32(const float* __restrict__ A,
                                                     const float* __restrict__ Bw,
                                                     float* __restrict__ Cout,
                                                     int M, int N, int K) {
  __shared__ float sA[BM * LDSS];
  __shared__ float sB[BN * LDSS];

  const int tid  = threadIdx.x;
  const int lane = tid & 31;
  const int wave = tid >> 5;
  const int wm   = wave >> 1;      // 0..3 -> 32 rows each
  const int wn   = wave & 1;       // 0..1 -> 64 cols each
  const int lm   = lane & 15;
  const int h    = lane >> 4;      // lane-half selects K pair {0,1} vs {2,3}

  const int gm = blockIdx.y * BM;
  const int gn = blockIdx.x * BN;

  v8f acc[2][4];
  #pragma unroll
  for (int i = 0; i < 2; ++i)
    #pragma unroll
    for (int j = 0; j < 4; ++j)
      acc[i][j] = (v8f)(0.0f);

  const int lr = tid >> 3;        // 0..31
  const int lc = (tid & 7) * 4;   // 0,4,...,28

  for (int kk = 0; kk < K; kk += BKT) {
    // cooperative loads: A tile (feat rows), B tile (weight rows == B^T columns)
    #pragma unroll
    for (int p = 0; p < 4; ++p) {
      int r   = lr + p * 32;
      int row = gm + r;
      float4 v = make_float4(0.f, 0.f, 0.f, 0.f);
      if (row < M) v = *(const float4*)&A[(size_t)row * K + kk + lc];
      *(float4*)&sA[r * LDSS + lc] = v;
    }
    #pragma unroll
    for (int p = 0; p < 4; ++p) {
      int r   = lr + p * 32;
      int row = gn + r;
      float4 v = make_float4(0.f, 0.f, 0.f, 0.f);
      if (row < N) v = *(const float4*)&Bw[(size_t)row * K + kk + lc];
      *(float4*)&sB[r * LDSS + lc] = v;
    }
    __syncthreads();

    if (kk + BKT < K) {
      __builtin_prefetch(&A[(size_t)(gm + lr) * K + kk + BKT + lc]);
      if (gn + lr < N) __builtin_prefetch(&Bw[(size_t)(gn + lr) * K + kk + BKT + lc]);
    }

    #pragma unroll
    for (int ks = 0; ks < BKT; ks += 4) {
      // A frag 16x4: lane<16 -> K=ks+{0,1}; lane>=16 -> K=ks+{2,3}; M = lm
      v2f a0 = *(const v2f*)&sA[(wm * 32 +      lm) * LDSS + ks + 2 * h];
      v2f a1 = *(const v2f*)&sA[(wm * 32 + 16 + lm) * LDSS + ks + 2 * h];
      v2f bfrag[4];
      #pragma unroll
      for (int j = 0; j < 4; ++j)
        bfrag[j] = *(const v2f*)&sB[(wn * 64 + j * 16 + lm) * LDSS + ks + 2 * h];
      #pragma unroll
      for (int j = 0; j < 4; ++j) {
        acc[0][j] = __builtin_amdgcn_wmma_f32_16x16x4_f32(
            false, a0, false, bfrag[j], (short)0, acc[0][j], false, false);
        acc[1][j] = __builtin_amdgcn_wmma_f32_16x16x4_f32(
            false, a1, false, bfrag[j], (short)0, acc[1][j], false, false);
      }
    }
    __syncthreads();
  }

  // epilogue: C/D layout -> vgpr v: row = base + v + 8*h, col = base_n + lm
  #pragma unroll
  for (int i = 0; i < 2; ++i) {
    int mbase = gm + wm * 32 + i * 16 + 8 * h;
    #pragma unroll
    for (int j = 0; j < 4; ++j) {
      int col = gn + wn * 64 + j * 16 + lm;
      if (col < N) {
        #pragma unroll
        for (int v = 0; v < 8; ++v) {
          int row = mbase + v;
          if (row < M) Cout[(size_t)row * N + col] = acc[i][j][v];
        }
      }
    }
  }
}

// ---------------------------------------------------------------------------
// Kernel 3: per-row reduction. One wave per row.
//  - labeled distance, top-10 smallest non-label distances (min + next 9)
//  - weights = 0.5*(sigmoid(100*(min-labeled)) + exp(100*(min-avg9)))
//  - normalize logits row in place.
// ---------------------------------------------------------------------------
__global__ __launch_bounds__(32) void rowreduce_kernel(float* __restrict__ logits,
                                                       const float* __restrict__ invf,
                                                       const float* __restrict__ invw,
                                                       const int* __restrict__ labels,
                                                       float* __restrict__ wts,
                                                       int N) {
  const int b    = blockIdx.x;
  const int lane = threadIdx.x;
  __shared__ float cand[32 * 10];
  __shared__ float sh_inv;

  float* row = logits + (size_t)b * N;
  const float INF = __builtin_inff();

  float t[10];
  #pragma unroll
  for (int i = 0; i < 10; ++i) t[i] = INF;

  float sumsq = 0.f, labeled = INF;
  const int   lab = labels[b];
  const float fi  = invf[b];

  for (int c = lane; c < N; c += 32) {
    float l = row[c];
    sumsq += l * l;
    float d = (1.0f - l * fi * invw[c]) * 0.5f;
    if (c == lab) {
      labeled = d;
    } else if (d < t[9]) {
      int p = 9;
      while (p > 0 && t[p - 1] > d) { t[p] = t[p - 1]; --p; }
      t[p] = d;
    }
  }
  #pragma unroll
  for (int off = 16; off > 0; off >>= 1) {
    sumsq  += __shfl_xor(sumsq, off, 32);
    labeled = fminf(labeled, __shfl_xor(labeled, off, 32));
  }
  #pragma unroll
  for (int i = 0; i < 10; ++i) cand[lane * 10 + i] = t[i];
  __syncthreads();

  if (lane == 0) {
    float best[10];
    #pragma unroll
    for (int i = 0; i < 10; ++i) best[i] = INF;
    for (int s = 0; s < 32; ++s) {
      for (int i = 0; i < 10; ++i) {
        float d = cand[s * 10 + i];
        if (d >= best[9]) break;   // per-lane lists are sorted ascending
        int p = 9;
        while (p > 0 && best[p - 1] > d) { best[p] = best[p - 1]; --p; }
        best[p] = d;
      }
    }
    float min_other = best[0];
    float s9 = 0.f;
    #pragma unroll
    for (int i = 1; i <= TOPK; ++i) s9 += best[i];
    float avg = s9 / (float)TOPK;
    float sig = 1.0f / (1.0f + __expf(-100.0f * (min_other - labeled)));
    float w   = (sig + __expf(100.0f * (min_other - avg))) * 0.5f;
    wts[b] = w;
    sh_inv = 1.0f / fmaxf(sqrtf(sumsq), EPSN);
  }
  __syncthreads();

  float inv = sh_inv;
  for (int c = lane; c < N; c += 32) row[c] *= inv;
}

// ---------------------------------------------------------------------------
// Kernel 4: descends = softmax(1 + cos(i*pi/200)) for i in 0..ep  (ep < 256)
// ---------------------------------------------------------------------------
__global__ __launch_bounds__(256) void descends_kernel(const int* __restrict__ ep_p,
                                                       float* __restrict__ desc) {
  __shared__ float red[256];
  const int ep = ep_p[0];
  const int i  = threadIdx.x;
  float e = 0.f;
  if (i <= ep)
    e = __expf(1.0f + __cosf((float)i * 3.14159265358979323846f / (float)TRAIN_EPOCH));
  red[i] = e;
  __syncthreads();
  #pragma unroll
  for (int off = 128; off > 0; off >>= 1) {
    if (i < off) red[i] += red[i + off];
    __syncthreads();
  }
  desc[i] = e / red[0];
}

// ---------------------------------------------------------------------------
// Kernels 5-7: deterministic last-writer-wins scatter (atomicMax on sample
// index is commutative -> deterministic under duplicate idx), then gather.
// ---------------------------------------------------------------------------
__global__ void initlast_kernel(const int* __restrict__ idx, int* __restrict__ lastb, int B) {
  int b = blockIdx.x * blockDim.x + threadIdx.x;
  if (b < B) lastb[idx[b]] = 0;
}

__global__ void scatmax_kernel(const int* __restrict__ idx, int* __restrict__ lastb, int B) {
  int b = blockIdx.x * blockDim.x + threadIdx.x;
  if (b < B) atomicMax(&lastb[idx[b]], b);
}

__global__ void finalw_kernel(const float* __restrict__ add_w,
                              const float* __restrict__ desc,
                              const float* __restrict__ wts,
                              const int* __restrict__ idx,
                              const int* __restrict__ lastb,
                              const int* __restrict__ ep_p,
                              float* __restrict__ out2,
                              int B, int TS) {
  int b = blockIdx.x * blockDim.x + threadIdx.x;
  if (b >= B) return;
  const int ep = ep_p[0];
  const int id = idx[b];
  float s = 0.f;
  for (int i = 0; i < ep; ++i) s += add_w[(size_t)i * TS + id] * desc[i];
  s += wts[lastb[id]] * desc[ep];
  out2[b] = s;
}

// ---------------------------------------------------------------------------
extern "C" void kernel_launch(void* const* d_in, const int* in_sizes, int n_in,
                              void* d_out, int out_size, void* d_ws, size_t ws_size,
                              hipStream_t stream) {
  const float* feat   = (const float*)d_in[0];
  const float* weight = (const float*)d_in[1];
  const float* add_w  = (const float*)d_in[2];
  const int*   labels = (const int*)d_in[3];
  const int*   idx    = (const int*)d_in[4];
  const int*   ep_p   = (const int*)d_in[5];

  const int B  = in_sizes[0] / FEAT_DIM;          // 8192
  const int C  = in_sizes[1] / FEAT_DIM;          // 1000
  const int TS = in_sizes[2] / TRAIN_EPOCH;       // 50000

  float* out_logits = (float*)d_out;
  float* out_w      = out_logits + (size_t)B * C;

  // workspace carve-up (all small)
  float* wts  = (float*)d_ws;                     // [B]
  float* invf = wts + B;                          // [B]
  float* invw = invf + B;                         // [C] (padded)
  float* desc = invw + ((C + 31) & ~31);          // [256]
  int*   lastb = (int*)(desc + 256);              // [TS]

  // 1) inverse norms
  {
    int rows = B + C;
    norms_kernel<<<(rows + 7) / 8, 256, 0, stream>>>(feat, weight, invf, invw, B, C);
  }
  // 2) WMMA f32 GEMM -> raw logits straight into d_out
  {
    dim3 grid((C + BN - 1) / BN, (B + BM - 1) / BM);
    gemm_wmma_f32<<<grid, 256, 0, stream>>>(feat, weight, out_logits, B, C, FEAT_DIM);
  }
  // 3) per-row: weights + in-place row normalization
  rowreduce_kernel<<<B, 32, 0, stream>>>(out_logits, invf, invw, labels, wts, C);
  // 4) cosine-annealed softmax coefficients
  descends_kernel<<<1, 256, 0, stream>>>(ep_p, desc);
  // 5-7) deterministic scatter + epoch-history gather
  initlast_kernel<<<(B + 255) / 256, 256, 0, stream>>>(idx, lastb, B);
  scatmax_kernel<<<(B + 255) / 256, 256, 0, stream>>>(idx, lastb, B);
  finalw_kernel<<<(B + 255) / 256, 256, 0, stream>>>(add_w, desc, wts, idx, lastb, ep_p,
                                                     out_w, B, TS);
}